// NLSPNModel_33363305955568
// MI455X (gfx1250) — compile-verified
//
#include <hip/hip_runtime.h>

// ---------------------------------------------------------------------------
// NLSPN propagation, MI455X (gfx1250, wave32).
// Precompute per-(pixel,neighbor) bilinear records once; zero-padded feature
// image makes the 6-iteration hot loop pure streaming + L2 gathers. Hot loop
// stages the record stream through LDS with CDNA5 async loads
// (global_load_async_to_lds_b128 / s_wait_asynccnt).
// ---------------------------------------------------------------------------

#define BATCH 4
#define H_ 240
#define W_ 1216
#define PAD 8
#define HP (H_ + 2 * PAD)          // 256
#define WP (W_ + 2 * PAD)          // 1232
#define HW (H_ * W_)               // 291840
#define NPIX (BATCH * HW)          // 1,167,360  (divisible by 256)
#define HPWP (HP * WP)             // 315,392
#define K9 9
#define PROP_TIME 6
#define EPS_SCALE 1e-8f
#define EPS_SUM 1e-4f
#define TB 256

// Toggle for the CDNA5 async-to-LDS staging path in the hot loop.
#ifndef USE_ASYNC_LDS
#define USE_ASYNC_LDS 1
#endif

// ---------------------------------------------------------------------------
// Shared helper: the 9 (idx, wx, wy, aff) records for pixel (b, y, x).
// Coordinates clamped so all 4 bilinear corners land inside the padded image;
// the zero pad ring reproduces the reference's valid-mask zero-padding exactly.
// ---------------------------------------------------------------------------
__device__ __forceinline__ void compute_recs(
    const float* __restrict__ aff_raw, const float* __restrict__ offset,
    float s, int b, int r, int y, int x,
    int* __restrict__ idx, float* __restrict__ wxv,
    float* __restrict__ wyv, float* __restrict__ affv)
{
    float t[8];
    float asum = 0.0f;
    const float sden = s + EPS_SCALE;
    const int base_a = b * 8 * HW + r;
#pragma unroll
    for (int n = 0; n < 8; ++n) {
        float a = tanhf(aff_raw[base_a + n * HW]) / sden;
        t[n] = a;
        asum += fabsf(a);
    }
    asum += EPS_SUM;
    asum = fmaxf(asum, 1.0f);
    float ssum = 0.0f;
#pragma unroll
    for (int n = 0; n < 8; ++n) {
        t[n] = t[n] / asum;
        ssum += t[n];
    }
    const float aref = 1.0f - ssum;

    const int base_o = b * 16 * HW + r;
#pragma unroll
    for (int k = 0; k < K9; ++k) {
        float oy, ox, a;
        if (k == 4) {
            oy = 0.0f; ox = 0.0f; a = aref;
        } else {
            const int n = (k < 4) ? k : (k - 1);
            oy = offset[base_o + (2 * n) * HW];
            ox = offset[base_o + (2 * n + 1) * HW];
            a = t[n];
        }
        float ys = (float)y + (float)(k / 3 - 1) + oy;
        float xs = (float)x + (float)(k % 3 - 1) + ox;
        ys = fminf(fmaxf(ys, -(float)PAD), (float)(H_ + PAD) - 1.5f);
        xs = fminf(fmaxf(xs, -(float)PAD), (float)(W_ + PAD) - 1.5f);
        const float y0 = floorf(ys);
        const float x0 = floorf(xs);
        idx[k]  = ((int)y0 + PAD) * WP + ((int)x0 + PAD);
        wyv[k]  = ys - y0;
        wxv[k]  = xs - x0;
        affv[k] = a;
    }
}

__device__ __forceinline__ float bilin_acc(const float* __restrict__ Pb,
                                           int i, float fx, float fy, float a)
{
    const float v00 = Pb[i];
    const float v01 = Pb[i + 1];
    const float v10 = Pb[i + WP];
    const float v11 = Pb[i + WP + 1];
    const float w00 = (1.0f - fy) * (1.0f - fx);
    const float w01 = (1.0f - fy) * fx;
    const float w10 = fy * (1.0f - fx);
    const float w11 = fy * fx;
    return a * (v00 * w00 + v01 * w01 + v10 * w10 + v11 * w11);
}

// ---------------------------------------------------------------------------
// Kernels
// ---------------------------------------------------------------------------
__global__ void k_zero(float* __restrict__ p, int n)
{
    const int i = blockIdx.x * blockDim.x + threadIdx.x;
    if (i < n) p[i] = 0.0f;
}

__global__ void k_pack(const float* __restrict__ pred,
                       const float* __restrict__ conf,
                       float* __restrict__ pad0)
{
    const int p = blockIdx.x * blockDim.x + threadIdx.x;
    if (p >= NPIX) return;
    const int b = p / HW;
    const int r = p - b * HW;
    const int y = r / W_;
    const int x = r - y * W_;
    pad0[(size_t)b * HPWP + (size_t)(y + PAD) * WP + (x + PAD)] =
        pred[p] * conf[p];
}

__global__ void k_precompute(const float* __restrict__ aff_raw,
                             const float* __restrict__ offset,
                             const float* __restrict__ sc,
                             float4* __restrict__ rec)
{
    const int p = blockIdx.x * blockDim.x + threadIdx.x;
    if (p >= NPIX) return;
    const int b = p / HW;
    const int r = p - b * HW;
    const int y = r / W_;
    const int x = r - y * W_;
    const float s = sc[0];

    int   idx[K9];
    float wx[K9], wy[K9], af[K9];
    compute_recs(aff_raw, offset, s, b, r, y, x, idx, wx, wy, af);

#pragma unroll
    for (int k = 0; k < K9; ++k) {
        float4 v;
        v.x = __int_as_float(idx[k]);
        v.y = wx[k];
        v.z = wy[k];
        v.w = af[k];
        rec[(size_t)k * NPIX + p] = v;   // plane-per-k: coalesced b128 stores
    }
}

// Hot-loop iteration (precomputed-record path). NPIX % TB == 0 -> all lanes
// of every block are active (EXEC all-ones).
__global__ void __launch_bounds__(TB)
k_iter(const float4* __restrict__ rec,
       const float* __restrict__ prev,
       float* __restrict__ next,
       const float* __restrict__ conf,
       const float* __restrict__ dep,
       float* __restrict__ outp)
{
    const int tid = threadIdx.x;
    const int p = blockIdx.x * TB + tid;
    const int b = p / HW;
    const int r = p - b * HW;
    const int y = r / W_;
    const int x = r - y * W_;

    // CDNA5 streaming prefetch on the record array (global_prefetch_b8).
    if (p + 16384 < NPIX)
        __builtin_prefetch((const void*)(rec + (size_t)8 * NPIX + p + 16384), 0, 0);

#if USE_ASYNC_LDS
    // Stage the 9 per-pixel records through LDS with async loads on ASYNCcnt:
    // each lane copies its own 16B record per plane (no cross-lane sharing,
    // so a single s_wait_asynccnt suffices; no barrier needed).
    __shared__ float4 srec[K9][TB];
    // Generic LDS address: low 32 bits are the wave-relative LDS byte offset.
    const unsigned int lds0 = (unsigned int)(unsigned long long)&srec[0][tid];
#pragma unroll
    for (int k = 0; k < K9; ++k) {
        const unsigned long long gaddr =
            (unsigned long long)(const void*)(rec + (size_t)k * NPIX + p);
        const unsigned int laddr = lds0 + (unsigned int)(k * (TB * 16));
        asm volatile("global_load_async_to_lds_b128 %0, %1, off"
                     :
                     : "v"(laddr), "v"(gaddr)
                     : "memory");
    }
    asm volatile("s_wait_asynccnt 0x0" ::: "memory");
#endif

    const float* Pb = prev + (size_t)b * HPWP;
    float acc = 0.0f;
#pragma unroll
    for (int k = 0; k < K9; ++k) {
#if USE_ASYNC_LDS
        const float4 v = srec[k][tid];                 // ds_load_b128
#else
        const float4 v = rec[(size_t)k * NPIX + p];    // global_load_b128
#endif
        acc += bilin_acc(Pb, __float_as_int(v.x), v.y, v.z, v.w);
    }
    const float d = dep[p];
    const float o = (d > 0.0f) ? d : acc;
    next[(size_t)b * HPWP + (size_t)(y + PAD) * WP + (x + PAD)] = o * conf[p];
    if (outp) outp[p] = o;
}

// Fallback iteration: recompute records on the fly (small-workspace path).
__global__ void k_iter_fused(const float* __restrict__ aff_raw,
                             const float* __restrict__ offset,
                             const float* __restrict__ sc,
                             const float* __restrict__ prev,
                             float* __restrict__ next,
                             const float* __restrict__ conf,
                             const float* __restrict__ dep,
                             float* __restrict__ outp)
{
    const int p = blockIdx.x * blockDim.x + threadIdx.x;
    if (p >= NPIX) return;
    const int b = p / HW;
    const int r = p - b * HW;
    const int y = r / W_;
    const int x = r - y * W_;
    const float s = sc[0];

    int   idx[K9];
    float wx[K9], wy[K9], af[K9];
    compute_recs(aff_raw, offset, s, b, r, y, x, idx, wx, wy, af);

    const float* Pb = prev + (size_t)b * HPWP;
    float acc = 0.0f;
#pragma unroll
    for (int k = 0; k < K9; ++k)
        acc += bilin_acc(Pb, idx[k], wx[k], wy[k], af[k]);

    const float d = dep[p];
    const float o = (d > 0.0f) ? d : acc;
    next[(size_t)b * HPWP + (size_t)(y + PAD) * WP + (x + PAD)] = o * conf[p];
    if (outp) outp[p] = o;
}

// ---------------------------------------------------------------------------
// Host launcher
// ---------------------------------------------------------------------------
extern "C" void kernel_launch(void* const* d_in, const int* in_sizes, int n_in,
                              void* d_out, int out_size, void* d_ws, size_t ws_size,
                              hipStream_t stream)
{
    (void)in_sizes; (void)n_in; (void)out_size;

    const float* aff_raw = (const float*)d_in[0];
    const float* offset  = (const float*)d_in[1];
    const float* conf    = (const float*)d_in[2];
    const float* pred    = (const float*)d_in[3];
    const float* dep     = (const float*)d_in[4];
    const float* sc      = (const float*)d_in[5];
    float* out = (float*)d_out;

    const size_t padElems = (size_t)BATCH * HPWP;          // 1,261,568 floats
    float* pad0 = (float*)d_ws;
    float* pad1 = pad0 + padElems;
    const size_t padBytes = 2ull * padElems * sizeof(float);
    float4* rec = (float4*)((char*)d_ws + padBytes);
    const size_t recBytes = (size_t)K9 * NPIX * sizeof(float4); // ~168 MB
    const bool big = (ws_size >= padBytes + recBytes);

    const int gPad = (int)((2 * padElems + TB - 1) / TB);
    const int gPix = (NPIX + TB - 1) / TB;                 // 4560, exact

    // Zero both padded buffers (pad ring must be 0 every call), then pack
    // ff_0 = pred_init * conf into the interior of pad0.
    k_zero<<<gPad, TB, 0, stream>>>(pad0, (int)(2 * padElems));
    k_pack<<<gPix, TB, 0, stream>>>(pred, conf, pad0);

    float* a = pad0;
    float* b = pad1;
    if (big) {
        k_precompute<<<gPix, TB, 0, stream>>>(aff_raw, offset, sc, rec);
        for (int t = 0; t < PROP_TIME; ++t) {
            float* op = (t == PROP_TIME - 1) ? out : nullptr;
            k_iter<<<gPix, TB, 0, stream>>>(rec, a, b, conf, dep, op);
            float* tmp = a; a = b; b = tmp;
        }
    } else {
        for (int t = 0; t < PROP_TIME; ++t) {
            float* op = (t == PROP_TIME - 1) ? out : nullptr;
            k_iter_fused<<<gPix, TB, 0, stream>>>(aff_raw, offset, sc,
                                                  a, b, conf, dep, op);
            float* tmp = a; a = b; b = tmp;
        }
    }
}